// RLLGMN_23158463660468
// MI455X (gfx1250) — compile-verified
//
#include <hip/hip_runtime.h>
#include <hip/hip_bf16.h>
#include <stddef.h>

// ---------------------------------------------------------------------------
// R-LLGMN forward for MI455X (gfx1250, wave32, WMMA).
//
// B=256, T=256, D=24, C=10, K=5, M=3, H=325 (padded to 328), N=750 -> m-major
// reordered/padded to 3 blocks of 256 cols (q = c*25+k*5+j in [0,250)).
//
// Phase 1: xnl expansion -> Xpad [B*T, 328] f32 (cols 325..327 zero)
// Phase 2: W[h, c,k,j,m] -> Wr[h, m*256 + q] f32, zero padded ([328, 768])
// Phase 3: f32 WMMA GEMM: logits[bt, m*256+q] = Xpad @ Wr, then per-lane
//          act = exp(l0)+exp(l1)+exp(l2) -> act [B*T, 256].
//          Each wave computes a 32x16 output tile (2 row-tiles x 3 m-blocks
//          = 6 accumulators) so the 6 B-fragment loads per K-step are
//          amortized over 6 WMMAs instead of 3.
// Phase 4: sequential scan over T with alpha[C,K] state per batch element.
// ---------------------------------------------------------------------------

#define Bsz 256
#define Tsz 256
#define Dsz 24
#define Csz 10
#define Ksz 5
#define Hpad 328              // 325 padded to multiple of 4
#define Npad 768              // 3 * 256
#define Qpad 256              // 250 padded to 256
#define NROWS (Bsz * Tsz)     // 65536

typedef __attribute__((ext_vector_type(2))) float v2f;
typedef __attribute__((ext_vector_type(8))) float v8f;

// ---------------- Phase 1: nonlinear expansion --------------------------
__global__ void __launch_bounds__(256)
k_expand(const float* __restrict__ x, float* __restrict__ xpad) {
  const int row = blockIdx.x;                 // bt = b*T + t
  __shared__ float xs[Dsz];
  const int tid = threadIdx.x;
  if (tid < Dsz) xs[tid] = x[(size_t)row * Dsz + tid];
  __syncthreads();
  for (int h = tid; h < Hpad; h += 256) {
    float v;
    if (h == 0)        v = 1.0f;
    else if (h < 25)   v = xs[h - 1];
    else if (h < 325) {
      int p = h - 25;                         // index into triu pairs (i<=j)
      int i = 0;
      while (p >= Dsz - i) { p -= Dsz - i; ++i; }
      v = xs[i] * xs[i + p];
    } else             v = 0.0f;              // K-padding
    xpad[(size_t)row * Hpad + h] = v;
  }
}

// ---------------- Phase 2: weight reorder/pad ---------------------------
// Wr[h, m*256 + q] = W[h*750 + q*3 + m], zeros elsewhere.
__global__ void __launch_bounds__(256)
k_wreorder(const float* __restrict__ w, float* __restrict__ wr) {
  const int idx = blockIdx.x * 256 + threadIdx.x;
  if (idx >= Hpad * Npad) return;
  const int h = idx / Npad;
  const int n = idx % Npad;
  const int m = n >> 8;
  const int q = n & 255;
  float v = 0.0f;
  if (h < 325 && q < 250) v = w[(size_t)h * 750 + q * 3 + m];
  wr[idx] = v;
}

// ---------------- Phase 3: f32 WMMA GEMM + exp + m-sum ------------------
// One wave per (32-row mpair, 16-col qtile); 6 accumulators
// = 2 row-tiles x 3 m-blocks.
__global__ void __launch_bounds__(256)
k_gemm(const float* __restrict__ xpad, const float* __restrict__ wr,
       float* __restrict__ act) {
  const int lane  = threadIdx.x & 31;
  const int wave  = blockIdx.x * 8 + (threadIdx.x >> 5);
  const int mpair = wave >> 4;                // 0..2047 (32 rows each)
  const int qtile = wave & 15;                // 0..15
  const int row0  = mpair << 5;
  const int n0    = qtile << 4;
  const int r     = lane & 15;
  const int half  = lane >> 4;                // 0: lanes 0-15, 1: lanes 16-31

  // A fragment source rows (ISA: 16x4 f32 A -> lanes0-15 hold K={0,1},
  // lanes16-31 hold K={2,3}; row = tile_row0 + (lane&15)).
  const float* xrow0 = xpad + (size_t)(row0 + r) * Hpad;
  const float* xrow1 = xrow0 + (size_t)16 * Hpad;

  v8f acc0a = {}, acc1a = {}, acc2a = {};     // rows row0..row0+15
  v8f acc0b = {}, acc1b = {}, acc2b = {};     // rows row0+16..row0+31
  for (int kk = 0; kk < Hpad; kk += 4) {
    v2f a0, a1;
    a0.x = xrow0[kk + 2 * half + 0];
    a0.y = xrow0[kk + 2 * half + 1];
    a1.x = xrow1[kk + 2 * half + 0];
    a1.y = xrow1[kk + 2 * half + 1];
    // B fragment (4x16): VGPR v holds row kk + v + 2*half, col n0 + r.
    const float* wb = wr + (size_t)(kk + 2 * half) * Npad + (n0 + r);
    v2f b0, b1, b2;
    b0.x = wb[0];          b0.y = wb[Npad];
    b1.x = wb[256];        b1.y = wb[256 + Npad];
    b2.x = wb[512];        b2.y = wb[512 + Npad];
    acc0a = __builtin_amdgcn_wmma_f32_16x16x4_f32(false, a0, false, b0,
                                                  (short)0, acc0a, false, false);
    acc1a = __builtin_amdgcn_wmma_f32_16x16x4_f32(false, a0, false, b1,
                                                  (short)0, acc1a, false, false);
    acc2a = __builtin_amdgcn_wmma_f32_16x16x4_f32(false, a0, false, b2,
                                                  (short)0, acc2a, false, false);
    acc0b = __builtin_amdgcn_wmma_f32_16x16x4_f32(false, a1, false, b0,
                                                  (short)0, acc0b, false, false);
    acc1b = __builtin_amdgcn_wmma_f32_16x16x4_f32(false, a1, false, b1,
                                                  (short)0, acc1b, false, false);
    acc2b = __builtin_amdgcn_wmma_f32_16x16x4_f32(false, a1, false, b2,
                                                  (short)0, acc2b, false, false);
  }

  // act[row, q] = exp(m0) + exp(m1) + exp(m2); D VGPR v -> row v + 8*half.
  float* outa = act + (size_t)(row0 + 8 * half) * Qpad + (n0 + r);
  float* outb = outa + (size_t)16 * Qpad;
#pragma unroll
  for (int v = 0; v < 8; ++v) {
    float sa = __expf(acc0a[v]) + __expf(acc1a[v]) + __expf(acc2a[v]);
    float sb = __expf(acc0b[v]) + __expf(acc1b[v]) + __expf(acc2b[v]);
    outa[(size_t)v * Qpad] = sa;
    outb[(size_t)v * Qpad] = sb;
  }
}

// ---------------- Phase 4: sequential alpha scan ------------------------
// One 64-thread block per batch b; thread l<50 owns state (c=l/5, j=l%5).
__global__ void __launch_bounds__(64)
k_scan(const float* __restrict__ act, float* __restrict__ out) {
  const int b = blockIdx.x;
  const int l = threadIdx.x;                  // 0..63
  const int c = l / Ksz;
  const int j = l % Ksz;
  __shared__ float alpha[64];
  __shared__ float red[64];
  alpha[l] = (l < Csz * Ksz) ? 1.0f : 0.0f;
  __syncthreads();

  const float* arow = act + (size_t)b * Tsz * Qpad;
  for (int t = 0; t < Tsz; ++t) {
    float s = 0.0f;
    if (l < Csz * Ksz) {
      const float* a = arow + (size_t)t * Qpad + c * 25 + j;
#pragma unroll
      for (int k = 0; k < Ksz; ++k)
        s += alpha[c * Ksz + k] * a[k * Ksz];  // act[c,k,j]
    }
    red[l] = s;                                // s==0 for l>=50
    __syncthreads();
    for (int st = 32; st > 0; st >>= 1) {      // tree reduce -> red[0]
      if (l < st) red[l] += red[l + st];
      __syncthreads();
    }
    const float norm = red[0];
    if (l < Csz * Ksz) alpha[l] = s / norm;
    __syncthreads();                           // orders alpha/red for next t
  }

  if (l < Csz * Ksz && j == 0) {               // out[b,c] = sum_k alpha[c,k]
    float o = 0.0f;
#pragma unroll
    for (int k = 0; k < Ksz; ++k) o += alpha[c * Ksz + k];
    out[(size_t)b * Csz + c] = o;
  }
}

// ---------------------------------------------------------------------------
extern "C" void kernel_launch(void* const* d_in, const int* in_sizes, int n_in,
                              void* d_out, int out_size, void* d_ws,
                              size_t ws_size, hipStream_t stream) {
  (void)in_sizes; (void)n_in; (void)out_size; (void)ws_size;
  const float* x = (const float*)d_in[0];   // [B,T,D]
  const float* w = (const float*)d_in[1];   // [H,C,K,K,M] == [325, 750]
  float* out = (float*)d_out;               // [B,C]

  char* ws = (char*)d_ws;
  const size_t xpad_bytes = (size_t)NROWS * Hpad * sizeof(float); // ~82 MB
  const size_t wr_bytes   = (size_t)Hpad * Npad * sizeof(float);  // ~1 MB
  float* xpad = (float*)ws;
  float* wr   = (float*)(ws + xpad_bytes);
  float* actb = (float*)(ws + xpad_bytes + wr_bytes);             // ~64 MB

  k_expand  <<<NROWS, 256, 0, stream>>>(x, xpad);
  k_wreorder<<<(Hpad * Npad + 255) / 256, 256, 0, stream>>>(w, wr);
  // 32768 waves = 2048 row-pairs x 16 qtiles; 8 waves (one row-pair group
  // shares its A rows in WGP$) per 256-thread block.
  k_gemm    <<<NROWS / 16, 256, 0, stream>>>(xpad, wr, actb);
  k_scan    <<<Bsz, 64, 0, stream>>>(actb, out);
}